// SinkhornDual_1571958030811
// MI455X (gfx1250) — compile-verified
//
#include <hip/hip_runtime.h>

// Sinkhorn dual objective + gradient, N = 8192, lambda = 1.
// Pass 1: column sums of exp(alpha_i - M_ij) via V_WMMA_F32_16X16X4_F32
//         (A = ones 16x4, B = 4x16 tile of E; accumulator carries column sums).
// Pass 2: row sums of exp(alpha_i + beta_j - M_ij), per-lane acc + shfl reduce.
// Memory bound: 2 x 256MB reads of M -> ~22us floor at 23.3 TB/s.

#define N_DIM 8192
constexpr float LAM = 1.0f;

typedef __attribute__((ext_vector_type(2))) float v2f;
typedef __attribute__((ext_vector_type(8))) float v8f;

__global__ void init_kernel(float* __restrict__ S, float* __restrict__ out0) {
    int i = blockIdx.x * blockDim.x + threadIdx.x;
    if (i < N_DIM) S[i] = 0.0f;
    if (i == 0) out0[0] = 0.0f;
}

// Each wave owns 16 columns (j0..j0+15) and a 512-row chunk (blockIdx.y).
// Per iteration it forms a 4x16 tile of E = exp(alpha - M) in the WMMA
// B-matrix layout and accumulates column sums with D = ones(16x4) x B + D.
__global__ __launch_bounds__(256) void colsum_kernel(const float* __restrict__ M,
                                                     const float* __restrict__ alpha,
                                                     float* __restrict__ S) {
    const int lane = threadIdx.x & 31;
    const int wave = threadIdx.x >> 5;
    const int half = lane >> 4;   // which half-wave (k = 2*half .. 2*half+1)
    const int ln   = lane & 15;   // column within tile
    const int colTile = blockIdx.x * 8 + wave;           // 0..511
    const int j = colTile * 16 + ln;
    const int rowBase = blockIdx.y * (N_DIM / 16);       // 512-row chunks

    v8f acc = {};
    v2f ones; ones.x = 1.0f; ones.y = 1.0f;              // A[m,k] = 1 for all lanes/VGPRs

    for (int it = 0; it < (N_DIM / 16) / 4; ++it) {      // 128 iterations of 4 rows
        const int i0 = rowBase + it * 4 + half * 2;
        const size_t base = (size_t)i0 * N_DIM + j;
        const float m0 = M[base];
        const float m1 = M[base + N_DIM];
        __builtin_prefetch(&M[base + (size_t)8 * N_DIM], 0, 1);
        v2f b;
        b.x = __expf(LAM * (alpha[i0]     - m0));
        b.y = __expf(LAM * (alpha[i0 + 1] - m1));
        acc = __builtin_amdgcn_wmma_f32_16x16x4_f32(
            /*neg_a=*/false, ones, /*neg_b=*/false, b,
            /*c_mod=*/(short)0, acc, /*reuse_a=*/false, /*reuse_b=*/false);
    }
    // D[m,n] = column-sum, replicated over m; lanes 0..15 hold columns j0..j0+15.
    if (half == 0) atomicAdd(&S[j], acc[0]);
}

__global__ void beta_kernel(const float* __restrict__ S,
                            const float* __restrict__ wb,
                            const float* __restrict__ alpha,
                            const float* __restrict__ wa,
                            float* __restrict__ beta,
                            float* __restrict__ out0) {
    int j = blockIdx.x * blockDim.x + threadIdx.x;
    if (j < N_DIM) {
        float b = (logf(wb[j]) - logf(S[j])) / LAM;
        beta[j] = b;
        // obj pieces: -alpha.wa - beta.wb
        atomicAdd(out0, -alpha[j] * wa[j] - b * wb[j]);
    }
}

// Each wave owns 4 rows; lane halves take row pairs; lanes sweep 16 columns
// per iteration (coalesced 64B per row per half-wave). Final 16-lane xor reduce.
__global__ __launch_bounds__(256) void rowsum_kernel(const float* __restrict__ M,
                                                     const float* __restrict__ alpha,
                                                     const float* __restrict__ beta,
                                                     const float* __restrict__ wa,
                                                     float* __restrict__ out) {
    const int lane = threadIdx.x & 31;
    const int wave = threadIdx.x >> 5;
    const int half = lane >> 4;
    const int ln   = lane & 15;
    const int rowTile = blockIdx.x * 8 + wave;           // 0..2047
    const int i0 = rowTile * 4 + half * 2;               // this lane's row pair
    const float a0 = alpha[i0];
    const float a1 = alpha[i0 + 1];

    float accx = 0.0f, accy = 0.0f;
    for (int it = 0; it < N_DIM / 16; ++it) {            // 512 iterations
        const int j = it * 16 + ln;
        const size_t base = (size_t)i0 * N_DIM + j;
        const float m0 = M[base];
        const float m1 = M[base + N_DIM];
        __builtin_prefetch(&M[base + 128], 0, 1);
        const float bj = beta[j];
        accx += __expf(LAM * (a0 + bj - m0));
        accy += __expf(LAM * (a1 + bj - m1));
    }
    // reduce over the 16-lane column dimension (stays within each half-wave)
    for (int mask = 8; mask >= 1; mask >>= 1) {
        accx += __shfl_xor(accx, mask, 32);
        accy += __shfl_xor(accy, mask, 32);
    }
    if (ln == 0) {
        out[1 + i0]     = accx - wa[i0];
        out[1 + i0 + 1] = accy - wa[i0 + 1];
        atomicAdd(&out[0], (accx + accy) / LAM);
    }
}

extern "C" void kernel_launch(void* const* d_in, const int* in_sizes, int n_in,
                              void* d_out, int out_size, void* d_ws, size_t ws_size,
                              hipStream_t stream) {
    const float* alpha = (const float*)d_in[0];
    const float* wa    = (const float*)d_in[1];
    const float* wb    = (const float*)d_in[2];
    const float* M     = (const float*)d_in[3];
    float* out  = (float*)d_out;            // out[0]=obj, out[1..8192]=grad
    float* S    = (float*)d_ws;             // N column sums
    float* beta = S + N_DIM;                // N betas

    init_kernel<<<N_DIM / 256, 256, 0, stream>>>(S, out);
    colsum_kernel<<<dim3(64, 16), 256, 0, stream>>>(M, alpha, S);
    beta_kernel<<<N_DIM / 256, 256, 0, stream>>>(S, wb, alpha, wa, beta, out);
    rowsum_kernel<<<256, 256, 0, stream>>>(M, alpha, beta, wa, out);
}